// RelationalAwareEncoder_9740985828004
// MI455X (gfx1250) — compile-verified
//
#include <hip/hip_runtime.h>
#include <hip/hip_bf16.h>

#define NN 200000
#define HH 50000
#define DD 64
#define N_LAYERS 2
#define LEAKY 0.2f
#define LN_EPS 1e-5f

typedef float v2f __attribute__((ext_vector_type(2)));
typedef float v8f __attribute__((ext_vector_type(8)));

// ---------------------------------------------------------------------------
// Zero-fill (float4 per thread)
// ---------------------------------------------------------------------------
__global__ __launch_bounds__(256) void fill_zero4(float4* __restrict__ p, size_t n4) {
    size_t i = (size_t)blockIdx.x * blockDim.x + threadIdx.x;
    if (i < n4) p[i] = make_float4(0.f, 0.f, 0.f, 0.f);
}

// ---------------------------------------------------------------------------
// Scatter SpMM: y[r_out[e]] += vals[e] * x[c_in[e]]   (rows of width 64)
// 32 threads per nonzero; each thread owns a float2 column pair.
// Native f32 atomic adds (device scope) via inline CDNA5 asm; the whole
// row scatter is two global_atomic_add_f32 per lane (L2-resident RMW).
// ---------------------------------------------------------------------------
__global__ __launch_bounds__(256) void spmm_scatter(const int* __restrict__ r_out,
                                                    const int* __restrict__ c_in,
                                                    const float* __restrict__ vals,
                                                    const float* __restrict__ x,
                                                    float* __restrict__ y,
                                                    int nnz) {
    unsigned long long tid = (unsigned long long)blockIdx.x * blockDim.x + threadIdx.x;
    int e = (int)(tid >> 5);
    if (e >= nnz) return;
    int p = ((int)tid & 31) << 1;           // column pair: 0,2,...,62

    int r = r_out[e];
    int c = c_in[e];
    float v = vals[e];

    const float2 xv = *reinterpret_cast<const float2*>(x + (size_t)c * DD + p);
    float a0 = v * xv.x;
    float a1 = v * xv.y;

    float* dst = y + (size_t)r * DD + p;
    asm volatile("global_atomic_add_f32 %0, %1, off scope:SCOPE_DEV"
                 :: "v"(dst), "v"(a0) : "memory");
    asm volatile("global_atomic_add_f32 %0, %1, off offset:4 scope:SCOPE_DEV"
                 :: "v"(dst), "v"(a1) : "memory");
    // s_endpgm performs an implicit wait-idle, so outstanding STOREcnt is safe.
}

// 3-level constant-index select: element i (0..7) of a v8f, per-lane i
__device__ __forceinline__ float pick8(const v8f v, int i) {
    float a0 = (i & 1) ? v[1] : v[0];
    float a1 = (i & 1) ? v[3] : v[2];
    float a2 = (i & 1) ? v[5] : v[4];
    float a3 = (i & 1) ? v[7] : v[6];
    float b0 = (i & 2) ? a1 : a0;
    float b1 = (i & 2) ? a3 : a2;
    return (i & 4) ? b1 : b0;
}

// ---------------------------------------------------------------------------
// Fused (leaky_relu) -> layernorm -> * gamma + beta -> + residual
// One wave32 per 16 rows. Row sums and row sums-of-squares are computed on
// the matrix pipe: acc(16x16) += A(16x4) * ones(4x16) over 16 K-chunks,
// i.e. V_WMMA_F32_16X16X4_F32. The A-matrix VGPR layout (lanes 0-15: K=0/1,
// lanes 16-31: K=2/3) is exactly a float2 load at col = 4c + (lane>>4)*2.
// Activations stay in VGPRs between the reduction and the normalize/store.
// ---------------------------------------------------------------------------
__global__ __launch_bounds__(256) void fused_ln(const float* __restrict__ h,
                                                const float* __restrict__ gamma,
                                                const float* __restrict__ beta,
                                                const float* __restrict__ residual,
                                                float* __restrict__ out,
                                                int nrows, int do_leaky) {
    int lane = threadIdx.x & 31;
    int wid  = threadIdx.x >> 5;
    int wrow = (blockIdx.x * (blockDim.x >> 5) + wid) * 16;   // wave-uniform
    if (wrow >= nrows) return;                                // uniform exit: EXEC stays all-ones

    int half = lane >> 4;     // 0: K pair {0,1}, 1: K pair {2,3}
    int l15  = lane & 15;     // row within tile

    const float* rowp = h + (size_t)(wrow + l15) * DD + half * 2;

    v8f csum = {};
    v8f csq  = {};
    v2f ones; ones[0] = 1.0f; ones[1] = 1.0f;
    float2 acts[16];

#pragma unroll
    for (int c = 0; c < 16; ++c) {
        float2 d = *reinterpret_cast<const float2*>(rowp + c * 4);
        if (do_leaky) {
            d.x = (d.x > 0.f) ? d.x : LEAKY * d.x;
            d.y = (d.y > 0.f) ? d.y : LEAKY * d.y;
        }
        acts[c] = d;
        v2f a;  a[0]  = d.x;        a[1]  = d.y;
        v2f a2; a2[0] = d.x * d.x;  a2[1] = d.y * d.y;
        csum = __builtin_amdgcn_wmma_f32_16x16x4_f32(false, a,  false, ones,
                                                     (short)0, csum, false, false);
        csq  = __builtin_amdgcn_wmma_f32_16x16x4_f32(false, a2, false, ones,
                                                     (short)0, csq,  false, false);
    }

    // C/D layout: VGPR j holds row j (lanes 0-15) / row j+8 (lanes 16-31).
    // Swap halves so each lane can select its own row's stats.
    v8f osum, osq;
#pragma unroll
    for (int k = 0; k < 8; ++k) {
        osum[k] = __shfl_xor(csum[k], 16, 32);
        osq[k]  = __shfl_xor(csq[k],  16, 32);
    }
    bool useOwn = (half == (l15 >> 3));
    int  idx    = l15 & 7;
    float mysum = useOwn ? pick8(csum, idx) : pick8(osum, idx);
    float mysq  = useOwn ? pick8(csq,  idx) : pick8(osq,  idx);

    float mu   = mysum * (1.0f / 64.0f);
    float ex2  = mysq  * (1.0f / 64.0f);
    float var  = ex2 - mu * mu;
    float rsig = rsqrtf(var + LN_EPS);

    const float* resp = residual + (size_t)(wrow + l15) * DD + half * 2;
    float*       outp = out      + (size_t)(wrow + l15) * DD + half * 2;

#pragma unroll
    for (int c = 0; c < 16; ++c) {
        int col  = c * 4 + half * 2;
        float2 g = *reinterpret_cast<const float2*>(gamma + col);
        float2 b = *reinterpret_cast<const float2*>(beta  + col);
        float2 r = *reinterpret_cast<const float2*>(resp + c * 4);
        float2 o;
        o.x = (acts[c].x - mu) * rsig * g.x + b.x + r.x;
        o.y = (acts[c].y - mu) * rsig * g.y + b.y + r.y;
        *reinterpret_cast<float2*>(outp + c * 4) = o;
    }
}

// ---------------------------------------------------------------------------
// Launcher
// ---------------------------------------------------------------------------
extern "C" void kernel_launch(void* const* d_in, const int* in_sizes, int n_in,
                              void* d_out, int out_size, void* d_ws, size_t ws_size,
                              hipStream_t stream) {
    const float* embs     = (const float*)d_in[0];
    const int*   A_rows   = (const int*)  d_in[1];
    const int*   A_cols   = (const int*)  d_in[2];
    const float* A_vals   = (const float*)d_in[3];
    const int*   S_rows   = (const int*)  d_in[4];
    const int*   S_cols   = (const int*)  d_in[5];
    const float* S_vals   = (const float*)d_in[6];
    const float* ln_gamma = (const float*)d_in[7];
    const float* ln_beta  = (const float*)d_in[8];
    float*       out      = (float*)d_out;

    const int nnz = in_sizes[1];

    float* bufA = (float*)d_ws;                       // [N, 64]
    float* bufB = bufA + (size_t)NN * DD;             // [N, 64]
    float* bufH = bufB + (size_t)NN * DD;             // [H, 64]

    auto zero = [&](float* p, size_t n) {
        size_t n4 = n >> 2;
        fill_zero4<<<(unsigned)((n4 + 255) / 256), 256, 0, stream>>>((float4*)p, n4);
    };
    unsigned spmm_blocks = (unsigned)(((unsigned long long)nnz * 32ull + 255ull) / 256ull);
    unsigned ln_blocks   = (NN + 127) / 128;          // 8 waves/block * 16 rows/wave

    const float* x = embs;
    for (int layer = 0; layer < N_LAYERS; ++layer) {
        // t0 = A^T @ x           -> bufA
        zero(bufA, (size_t)NN * DD);
        spmm_scatter<<<spmm_blocks, 256, 0, stream>>>(A_cols, A_rows, A_vals, x, bufA, nnz);
        // tH = S^T @ t0          -> bufH
        zero(bufH, (size_t)HH * DD);
        spmm_scatter<<<spmm_blocks, 256, 0, stream>>>(S_cols, S_rows, S_vals, bufA, bufH, nnz);
        // t2 = S @ tH            -> bufB   (x, if it was bufB, is dead by now)
        zero(bufB, (size_t)NN * DD);
        spmm_scatter<<<spmm_blocks, 256, 0, stream>>>(S_rows, S_cols, S_vals, bufH, bufB, nnz);
        // h  = A @ t2            -> bufA
        zero(bufA, (size_t)NN * DD);
        spmm_scatter<<<spmm_blocks, 256, 0, stream>>>(A_rows, A_cols, A_vals, bufB, bufA, nnz);
        // x_next = LN(act(h)) * gamma + beta + embs
        float* xn = (layer == N_LAYERS - 1) ? out : bufB;
        fused_ln<<<ln_blocks, 256, 0, stream>>>(bufA, ln_gamma + layer * DD,
                                                ln_beta + layer * DD, embs, xn,
                                                NN, (layer != N_LAYERS - 1) ? 1 : 0);
        x = xn;
    }
}